// GraphAttentionLayer_9096740733359
// MI455X (gfx1250) — compile-verified
//
#include <hip/hip_runtime.h>
#include <math.h>

typedef float v2f __attribute__((ext_vector_type(2)));
typedef float v4f __attribute__((ext_vector_type(4)));
typedef float v8f __attribute__((ext_vector_type(8)));

constexpr int H = 128;          // hidden dim
constexpr int LSEQ = 512;       // sequence length
constexpr int TILE = 16;        // positions per wave (WMMA N dim)
constexpr int WAVES = 4;        // waves per block
constexpr int ROWSTRIDE = 132;  // padded LDS row stride (floats) -> bank-conflict free
constexpr int KSTEPS = H / 4;   // 32 WMMA K-steps

__global__ __launch_bounds__(WAVES * 32)
void gat_wmma_kernel(const float* __restrict__ emb,
                     const float* __restrict__ a,
                     const int*   __restrict__ seq,
                     const int*   __restrict__ seq_l,
                     float*       __restrict__ out)
{
    __shared__ float lds[WAVES * TILE * ROWSTRIDE];
    __shared__ v2f   atab[KSTEPS * 32];   // pre-formed A fragments, shared by all waves

    const int lane   = threadIdx.x & 31;
    const int wave   = threadIdx.x >> 5;
    const int waveId = blockIdx.x * WAVES + wave;
    const int b      = waveId >> 5;            // L/TILE = 32 tiles per row
    const int tile   = waveId & 31;
    const int l0     = tile * TILE;

    float* xs = lds + wave * TILE * ROWSTRIDE;

    const int n    = lane & 15;                // position / M index within half-wave
    const int half = lane >> 4;

    // ---- Build A-fragment table once per block ----
    // A (16x4/step): row 0 = a1, row 1 = a2, rows 2..15 = 0
    //   lane layout: lanes 0-15 -> M=lane, K={0,1}; lanes 16-31 -> M=lane-16, K={2,3}
    for (int e = threadIdx.x; e < KSTEPS * 32; e += WAVES * 32) {
        const int kc = e >> 5;
        const int l2 = e & 31;
        const int ma = l2 & 15;
        const int k  = 4 * kc + 2 * (l2 >> 4);
        v2f v = {0.f, 0.f};
        if (ma == 0)      v = *(const v2f*)(a + k);
        else if (ma == 1) v = *(const v2f*)(a + H + k);
        atab[e] = v;
    }
    __syncthreads();

    const int sl = seq_l[b];

    // ---- Fill phase: stage 16 raw embedding rows into LDS (coalesced 512B/row) ----
    const int* seqRow = seq + (size_t)b * LSEQ + l0;
    for (int r = 0; r < TILE; ++r) {
        int idx = seqRow[r];
        v4f ev = *(const v4f*)(emb + (size_t)idx * H + 4 * lane);
        *(v4f*)(xs + r * ROWSTRIDE + 4 * lane) = ev;
    }

    // ---- Halo (position l0+16): x_next for m==15 and its dot with a2 ----
    v4f   xh4    = {0.f, 0.f, 0.f, 0.f};
    float d2halo = 0.f;
    if (l0 + TILE < LSEQ) {
        int idxh = seqRow[TILE];
        v4f ev = *(const v4f*)(emb + (size_t)idxh * H + 4 * lane);
        float ph = (float)(sl - (l0 + TILE));
        xh4 = ev + ph;
        v4f av = *(const v4f*)(a + H + 4 * lane);
        d2halo = xh4.x * av.x + xh4.y * av.y + xh4.z * av.z + xh4.w * av.w;
    }
    #pragma unroll
    for (int off = 16; off >= 1; off >>= 1)
        d2halo += __shfl_xor(d2halo, off, 32);

    // ---- Score phase via V_WMMA_F32_16X16X4_F32 ----
    // B (4x16/step): B[k][n] = x[n][4*kc+k]
    //   lane layout: lanes 0-15 -> K={0,1} (VGPR0/1), lanes 16-31 -> K={2,3}
    // D: lane n (0-15): c[0] = dot(a1, x[n]), c[1] = dot(a2, x[n])
    const float posn = (float)(sl - (l0 + n));

    v8f c = {0.f, 0.f, 0.f, 0.f, 0.f, 0.f, 0.f, 0.f};
    #pragma unroll
    for (int kc = 0; kc < KSTEPS; ++kc) {
        v2f afrag = atab[kc * 32 + lane];
        v2f ev = *(const v2f*)(xs + n * ROWSTRIDE + 4 * kc + 2 * half);
        v2f bfrag;
        bfrag.x = ev.x + posn;
        bfrag.y = ev.y + posn;
        c = __builtin_amdgcn_wmma_f32_16x16x4_f32(false, afrag, false, bfrag,
                                                  (short)0, c, false, false);
    }
    const float c0 = c[0];   // d1[n] on lanes 0..15
    const float c1 = c[1];   // d2[n] on lanes 0..15

    // ---- Output phase: per position, softmax over {s0,s1}, blend, mask ----
    const size_t outBase = ((size_t)b * LSEQ + l0) * H;
    #pragma unroll
    for (int m = 0; m < TILE; ++m) {
        float d1m = __shfl(c0, m, 32);
        float d2m = __shfl(c1, m, 32);
        float d2n = (m == TILE - 1) ? d2halo : __shfl(c1, m + 1, 32);

        float s0 = d1m + d2m;
        float s1 = d1m + d2n;
        float mx = fmaxf(s0, s1);
        float e0 = expf(s0 - mx);
        float e1 = expf(s1 - mx);
        float inv = 1.f / (e0 + e1);
        float att0 = e0 * inv;
        float att1 = e1 * inv;

        float posm = (float)(sl - (l0 + m));
        v4f er = *(const v4f*)(xs + m * ROWSTRIDE + 4 * lane);  // raw e
        v4f xm = er + posm;
        v4f xn;
        if (m == TILE - 1) {
            xn = xh4;
        } else {
            v4f ern = *(const v4f*)(xs + (m + 1) * ROWSTRIDE + 4 * lane);
            xn = ern + (posm - 1.f);
        }
        v4f u = att0 * xm + att1 * xn;
        bool valid = (l0 + m) < (sl - 1);
        v4f res = valid ? u : er;
        *(v4f*)(out + outBase + (size_t)m * H + 4 * lane) = res;
    }
}

extern "C" void kernel_launch(void* const* d_in, const int* in_sizes, int n_in,
                              void* d_out, int out_size, void* d_ws, size_t ws_size,
                              hipStream_t stream) {
    const float* emb   = (const float*)d_in[0];
    const float* a     = (const float*)d_in[1];
    const int*   seq   = (const int*)  d_in[2];
    const int*   seq_l = (const int*)  d_in[3];
    float*       out   = (float*)d_out;

    const int B = in_sizes[3];                      // 1024
    const int tiles = B * (LSEQ / TILE);            // 32768 waves
    const int blocks = tiles / WAVES;               // 8192 blocks of 4 waves

    gat_wmma_kernel<<<dim3(blocks), dim3(WAVES * 32), 0, stream>>>(
        emb, a, seq, seq_l, out);
}